// Pooling_90546500535080
// MI455X (gfx1250) — compile-verified
//
#include <hip/hip_runtime.h>
#include <stdint.h>

typedef float f4 __attribute__((ext_vector_type(4)));
typedef unsigned int u32x4 __attribute__((ext_vector_type(4)));
typedef int i32x4 __attribute__((ext_vector_type(4)));
typedef int i32x8 __attribute__((ext_vector_type(8)));

#define T_      8192
#define D_      1024
#define TILE_T  32
#define ROWS    40            // t0-3 .. t0+TILE_T+4  (stencil halo)
#define TILE_D  256           // floats per block along D
#define ROW_F4  (TILE_D / 4)  // 64 x 16B chunks per staged row
#define ROW_B   (TILE_D * 4)  // 1024 bytes per staged row
#define THREADS 256

#if __has_builtin(__builtin_amdgcn_tensor_load_to_lds) && \
    __has_builtin(__builtin_amdgcn_s_wait_tensorcnt)
#define HAVE_TDM 1
#else
#define HAVE_TDM 0
#endif

// Fallback gfx1250 async global->LDS copy (ASYNCcnt path).
__device__ __forceinline__ void async_copy16(const float* g, f4* l) {
    unsigned loff = (unsigned)(uintptr_t)(__attribute__((address_space(3))) f4*)l;
    unsigned long long ga = (unsigned long long)(uintptr_t)g;
    asm volatile("global_load_async_to_lds_b128 %0, %1, off"
                 :: "v"(loff), "v"(ga)
                 : "memory");
}
__device__ __forceinline__ void async_wait0() {
    asm volatile("s_wait_asynccnt 0" ::: "memory");
}

__global__ __launch_bounds__(THREADS)
void pool_kernel(const float* __restrict__ x, float* __restrict__ out) {
    __shared__ f4 tile[ROWS * ROW_F4];   // 40 KB

    const int tid   = threadIdx.x;
    const int d0q   = blockIdx.x * ROW_F4;   // column base, f4 units
    const int t0    = blockIdx.y * TILE_T;
    const int rbase = t0 - 3;

#if HAVE_TDM
    // ---- one TDM descriptor moves the whole 2D tile (40 x 1KB, 4KB pitch) ----
    if (tid < 32) {                       // single wave issues the DMA
        int rstart = rbase < 0 ? 0 : rbase;
        int ldsrow = rstart - rbase;      // skip rows that are never read (t0==0)
        int nrows  = ROWS - ldsrow;
        if (rstart + nrows > T_) nrows = T_ - rstart;   // stay inside allocation

        unsigned ldsa = (unsigned)(uintptr_t)(__attribute__((address_space(3))) f4*)&tile[0]
                      + (unsigned)(ldsrow * ROW_B);
        unsigned long long ga = (unsigned long long)(uintptr_t)
                                (x + (size_t)rstart * D_ + (size_t)d0q * 4);

        // D# group0: count=1 | lds_addr | global_addr[56:0] | type=2 ("image")
        u32x4 g0 = { 1u,
                     ldsa,
                     (unsigned)ga,
                     (unsigned)(ga >> 32) | (2u << 30) };
        // D# group1: mask=0, data_size=4B(2), tensor_dim0=1024, tensor_dim1=8192,
        //            tile_dim0=256, tile_dim1=nrows, tile_dim2=0, dim0_stride=1024
        i32x8 g1 = { 0x00020000,          // [17:16] data_size=2 (4 bytes)
                     0x04000000,          // [31:16]=tensor_dim0.lo16 = 1024
                     0x20000000,          // [15:0]=dim0.hi16=0, [31:16]=dim1.lo16=8192
                     0x01000000,          // [15:0]=dim1.hi16=0, [31:16]=tile_dim0=256
                     nrows,               // [15:0]=tile_dim1, [31:16]=tile_dim2=0
                     1024,                // tensor_dim0_stride lo32 = 1024 elements
                     0, 0 };              // stride hi / dim1_stride unused (2D)
        i32x4 z4 = { 0, 0, 0, 0 };
        i32x8 z8 = { 0, 0, 0, 0, 0, 0, 0, 0 };
        __builtin_amdgcn_tensor_load_to_lds(g0, g1, z4, z4, z8, 0);
        __builtin_amdgcn_s_wait_tensorcnt(0);   // TENSORcnt -> 0 before releasing WG
    }
#else
    // ---- fallback: per-lane async b128 copies (10 per thread), ASYNCcnt ----
    for (int c = tid; c < ROWS * ROW_F4; c += THREADS) {
        int li  = c >> 6;
        int col = c & 63;
        int r   = rbase + li;
        r = r < 0 ? 0 : (r > T_ - 1 ? T_ - 1 : r);
        const float* gp = x + (size_t)r * D_ + (size_t)(d0q + col) * 4;
        async_copy16(gp, &tile[c]);
    }
    async_wait0();
#endif
    __syncthreads();

    const int    DQ  = D_ / 4;               // 256 f4 per [T,D] row
    const size_t TDQ = (size_t)T_ * DQ;      // one [T,D] plane, in f4
    f4* o = (f4*)out;
    const f4 z = {0.f, 0.f, 0.f, 0.f};
    const float c2 = 0.70710678118654752440f;

    // ---- 32 t x 64 f4-columns; wave-uniform t => scalar masks, coalesced stores ----
    for (int s = tid; s < TILE_T * ROW_F4; s += THREADS) {
        int tl  = s >> 6;
        int col = s & 63;
        int t   = t0 + tl;

        // FFT window: rows fs..fs+3, valid iff row <= fe  (left-aligned zero pad)
        int fs = t - 1; if (fs < 0) fs = 0;
        int fe = t + 2; if (fe > T_ - 1) fe = T_ - 1;
        int fb = (fs - rbase) * ROW_F4 + col;
        f4 w0 = tile[fb];
        f4 w1 = (fs + 1 <= fe) ? tile[fb + 1 * ROW_F4] : z;
        f4 w2 = (fs + 2 <= fe) ? tile[fb + 2 * ROW_F4] : z;
        f4 w3 = (fs + 3 <= fe) ? tile[fb + 3 * ROW_F4] : z;

        // DWT window: rows qs..qs+7, valid iff row <= qe
        int qs = t - 3; if (qs < 0) qs = 0;
        int qe = t + 4; if (qe > T_ - 1) qe = T_ - 1;
        int qb = (qs - rbase) * ROW_F4 + col;
        f4 u0 = tile[qb];
        f4 u1 = (qs + 1 <= qe) ? tile[qb + 1 * ROW_F4] : z;
        f4 u2 = (qs + 2 <= qe) ? tile[qb + 2 * ROW_F4] : z;
        f4 u3 = (qs + 3 <= qe) ? tile[qb + 3 * ROW_F4] : z;
        f4 u4 = (qs + 4 <= qe) ? tile[qb + 4 * ROW_F4] : z;
        f4 u5 = (qs + 5 <= qe) ? tile[qb + 5 * ROW_F4] : z;
        f4 u6 = (qs + 6 <= qe) ? tile[qb + 6 * ROW_F4] : z;
        f4 u7 = (qs + 7 <= qe) ? tile[qb + 7 * ROW_F4] : z;

        // DFT-4 real part: C = {0,+-1} -> pure adds. coarse_fft mean telescopes to w0.
        f4 re0 = (w0 + w1) + (w2 + w3);
        f4 re1 = w0 - w2;                 // re3 == re1
        f4 re2 = (w0 - w1) + (w2 - w3);

        // Haar cA + mean
        f4 a0 = (u0 + u1) * c2;
        f4 a1 = (u2 + u3) * c2;
        f4 a2 = (u4 + u5) * c2;
        f4 a3 = (u6 + u7) * c2;
        f4 cd = ((a0 + a1) + (a2 + a3)) * 0.25f;

        // Output layout: [coarse_fft TD | fine_fft 4TD | coarse_dwt TD | fine_dwt 4TD]
        size_t rowq  = (size_t)t * DQ + d0q + col;            // [T,D]  plane index
        size_t rowq4 = (size_t)t * (4 * DQ) + d0q + col;      // [T,4D] plane index

        __builtin_nontemporal_store(w0,  o + rowq);                         // coarse_fft
        __builtin_nontemporal_store(re0, o + TDQ + rowq4);                  // fine_fft k=0
        __builtin_nontemporal_store(re1, o + TDQ + rowq4 + 1 * DQ);         // k=1
        __builtin_nontemporal_store(re2, o + TDQ + rowq4 + 2 * DQ);         // k=2
        __builtin_nontemporal_store(re1, o + TDQ + rowq4 + 3 * DQ);         // k=3
        __builtin_nontemporal_store(cd,  o + 5 * TDQ + rowq);               // coarse_dwt
        __builtin_nontemporal_store(a0,  o + 6 * TDQ + rowq4);              // fine_dwt m=0
        __builtin_nontemporal_store(a1,  o + 6 * TDQ + rowq4 + 1 * DQ);
        __builtin_nontemporal_store(a2,  o + 6 * TDQ + rowq4 + 2 * DQ);
        __builtin_nontemporal_store(a3,  o + 6 * TDQ + rowq4 + 3 * DQ);
    }
}

extern "C" void kernel_launch(void* const* d_in, const int* in_sizes, int n_in,
                              void* d_out, int out_size, void* d_ws, size_t ws_size,
                              hipStream_t stream) {
    (void)in_sizes; (void)n_in; (void)out_size; (void)d_ws; (void)ws_size;
    const float* x = (const float*)d_in[0];
    float* out = (float*)d_out;
    dim3 grid(D_ / TILE_D, T_ / TILE_T);   // 4 x 256 blocks
    pool_kernel<<<grid, THREADS, 0, stream>>>(x, out);
}